// CA_85332410237583
// MI455X (gfx1250) — compile-verified
//
#include <hip/hip_runtime.h>

typedef float v2f __attribute__((ext_vector_type(2)));
typedef float v8f __attribute__((ext_vector_type(8)));

#define PV1F 16777216.0f           // floor(2^31/128) == 2^24 (exact)
#define INV_PV1F (1.0f/16777216.0f)
#define PV2F 671088.0f             // floor(2^31/(25*128))

// workspace layout (float offsets)
#define OFF_K   0       // 800  : symmetrized conv kernels, layout [c][i][j][f]
#define OFF_W1  800     // 128  : W1 [f][o]
#define OFF_A2  1024    // 4096 : W2 A-tiles, [f][m][kg][lane][2]
#define OFF_A3  5120    // 4096 : W3 A-tiles
#define OFF_A4  9216    // 2048 : W4 A-tiles (rows 8..15 zero-padded)

#if __has_builtin(__builtin_amdgcn_tanhf)
#define TANHF(x) __builtin_amdgcn_tanhf(x)
#else
#define TANHF(x) tanhf(x)
#endif

__device__ __forceinline__ float qtanh(float x) {
  // floor(tanh(floor(x)/PV1)*PV1)  -- division by 2^24 is an exact multiply
  return floorf(TANHF(floorf(x) * INV_PV1F) * PV1F);
}

// ---------------------------------------------------------------------------
// Prep: symmetrize conv kernels + swizzle MLP weights into WMMA A-tile layout.
// A-tile (16x4 f32): lane l holds M = l&15; lanes 0-15 carry K=(4kg+0,4kg+1),
// lanes 16-31 carry K=(4kg+2,4kg+3), two consecutive floats per lane.
// ---------------------------------------------------------------------------
__global__ void ca_prep(const float* __restrict__ kernels,
                        const float* __restrict__ W1,
                        const float* __restrict__ W2,
                        const float* __restrict__ W3,
                        const float* __restrict__ W4,
                        float* __restrict__ ws) {
  const int t = threadIdx.x;

  // totalistic(kernels): z = 0.125*(sym(k)+sym(k^T)) - mean; mean == sum(k)/25
  for (int e = t; e < 800; e += 256) {
    int f = e & 3; int r = e >> 2;          // r = (c*5+i)*5+j
    int j = r % 5; int i = (r / 5) % 5; int c = r / 25;
    const float* kk = kernels + (f * 8 + c) * 25;
    float s = 0.0f;
#pragma unroll
    for (int q = 0; q < 25; ++q) s += kk[q];
    float a  = kk[i*5+j] + kk[(4-i)*5+j] + kk[i*5+(4-j)] + kk[(4-i)*5+(4-j)];
    float bt = kk[j*5+i] + kk[(4-j)*5+i] + kk[j*5+(4-i)] + kk[(4-j)*5+(4-i)];
    ws[OFF_K + e] = 0.125f * (a + bt) - s * (1.0f / 25.0f);
  }

  for (int e = t; e < 128; e += 256) ws[OFF_W1 + e] = W1[e];   // (f,o,1)->[f*32+o]

  // W2/W3: 4 f * 2 mtiles * 8 kg tiles, 64 floats per tile
  for (int e = t; e < 4096; e += 256) {
    int comp = e & 1; int l = (e >> 1) & 31; int tt = e >> 6;
    int kg = tt & 7; int m = (tt >> 3) & 1; int f = tt >> 4;
    int M = m * 16 + (l & 15);
    int K = kg * 4 + ((l >> 4) << 1) + comp;
    ws[OFF_A2 + e] = W2[(f * 32 + M) * 32 + K];
    ws[OFF_A3 + e] = W3[(f * 32 + M) * 32 + K];
  }

  // W4 (8x32) zero-padded to 16 rows: 4 f * 8 kg tiles
  for (int e = t; e < 2048; e += 256) {
    int comp = e & 1; int l = (e >> 1) & 31; int tt = e >> 6;
    int kg = tt & 7; int f = tt >> 3;
    int M = l & 15;
    int K = kg * 4 + ((l >> 4) << 1) + comp;
    ws[OFF_A4 + e] = (M < 8) ? W4[(f * 8 + M) * 32 + K] : 0.0f;
  }
}

// ---------------------------------------------------------------------------
// Main kernel: one block = 16x8 pixel tile of one batch image; 8 waves,
// wave w owns tile row w (16 pixels = WMMA N dimension).
// ---------------------------------------------------------------------------
__global__ __launch_bounds__(256)
void ca_main(const float* __restrict__ xin,
             const float* __restrict__ biases,
             const float* __restrict__ ws,
             const int* __restrict__ urate,
             float* __restrict__ out) {
  __shared__ float zt[8 * 12 * 20];   // wrap-padded quantized input tile
  __shared__ float kt[800];           // conv kernels [c][i][j][f]
  __shared__ float w1s[128];          // W1 [f][o]
  __shared__ float stage[8][512];     // per-wave 32x16 relayout buffer

  const int tid  = threadIdx.x;
  const int wave = tid >> 5;
  const int lane = tid & 31;
  const int b   = blockIdx.z;
  const int ty0 = blockIdx.y * 8;
  const int tx0 = blockIdx.x * 16;

  for (int i = tid; i < 800; i += 256) kt[i]  = ws[OFF_K + i];
  for (int i = tid; i < 128; i += 256) w1s[i] = ws[OFF_W1 + i];
  for (int i = tid; i < 1920; i += 256) {
    int c = i / 240; int r = i - c * 240; int y = r / 20; int x = r - y * 20;
    int gy = (ty0 + y - 2) & 255;     // wrap (H = 256)
    int gx = (tx0 + x - 2) & 255;
    zt[i] = floorf(xin[(((b * 8 + c) << 8) + gy) * 256 + gx] * PV2F);
  }
  __syncthreads();

  // ---- conv: both half-waves compute the same pixel (py, px) --------------
  const int py = wave;
  const int px = lane & 15;
  float cf0 = biases[0], cf1 = biases[1], cf2 = biases[2], cf3 = biases[3];
  for (int c = 0; c < 8; ++c) {
#pragma unroll
    for (int i = 0; i < 5; ++i) {
      int zb = (c * 12 + py + i) * 20 + px;
#pragma unroll
      for (int j = 0; j < 5; ++j) {
        float zv = zt[zb + j];
        const float4 kv = *(const float4*)&kt[((c * 5 + i) * 5 + j) * 4];
        cf0 += kv.x * zv; cf1 += kv.y * zv; cf2 += kv.z * zv; cf3 += kv.w * zv;
      }
    }
  }
  // p = floor(conv+bias)/PV2 ; layer input = floor(p*PV1)
  float ph[4];
  ph[0] = floorf((floorf(cf0) / PV2F) * PV1F);
  ph[1] = floorf((floorf(cf1) / PV2F) * PV1F);
  ph[2] = floorf((floorf(cf2) / PV2F) * PV1F);
  ph[3] = floorf((floorf(cf3) / PV2F) * PV1F);

  const int kksel = (lane >> 4) << 1;   // 0 (lanes 0-15) or 2 (lanes 16-31)
  const int n = lane & 15;
  float* mystage = stage[wave];

  v8f val[4];

#pragma unroll
  for (int f = 0; f < 4; ++f) {
    // ---- layer 1 (1->32): build B tiles directly in WMMA B layout ---------
    v2f B[8];
#pragma unroll
    for (int kg = 0; kg < 8; ++kg) {
      int k0 = f * 32 + kg * 4 + kksel;
      B[kg].x = qtanh(w1s[k0]     * ph[f]);
      B[kg].y = qtanh(w1s[k0 + 1] * ph[f]);
    }

    // ---- layers 2,3 (32->32) via V_WMMA_F32_16X16X4_F32 -------------------
#pragma unroll
    for (int layer = 0; layer < 2; ++layer) {
      const float* A = ws + (layer == 0 ? OFF_A2 : OFF_A3) + f * 1024;
      v2f a[16];
#pragma unroll
      for (int tnum = 0; tnum < 16; ++tnum)
        a[tnum] = *(const v2f*)&A[tnum * 64 + lane * 2];
      v8f acc0 = {};
      v8f acc1 = {};
#pragma unroll
      for (int kg = 0; kg < 8; ++kg) {
        acc0 = __builtin_amdgcn_wmma_f32_16x16x4_f32(
            false, a[kg],     false, B[kg], (short)0, acc0, false, false);
        acc1 = __builtin_amdgcn_wmma_f32_16x16x4_f32(
            false, a[8 + kg], false, B[kg], (short)0, acc1, false, false);
      }
      // elementwise quantized tanh, then D-layout -> B-layout through LDS
#pragma unroll
      for (int j = 0; j < 8; ++j) {
        int o = j + ((lane >> 4) << 3);           // rows j / j+8 per half-wave
        mystage[o * 16 + n]        = qtanh(acc0[j]);
        mystage[(o + 16) * 16 + n] = qtanh(acc1[j]);
      }
      __syncthreads();
#pragma unroll
      for (int kg = 0; kg < 8; ++kg) {
        int k0 = kg * 4 + kksel;
        B[kg].x = mystage[k0 * 16 + n];
        B[kg].y = mystage[(k0 + 1) * 16 + n];
      }
      __syncthreads();
    }

    // ---- layer 4 (32->8, rows 8..15 of A are zero padding) ----------------
    const float* A4 = ws + OFF_A4 + f * 512;
    v2f a4[8];
#pragma unroll
    for (int tnum = 0; tnum < 8; ++tnum)
      a4[tnum] = *(const v2f*)&A4[tnum * 64 + lane * 2];
    v8f acc = {};
#pragma unroll
    for (int kg = 0; kg < 8; ++kg)
      acc = __builtin_amdgcn_wmma_f32_16x16x4_f32(
          false, a4[kg], false, B[kg], (short)0, acc, false, false);
#pragma unroll
    for (int j = 0; j < 8; ++j)
      val[f][j] = TANHF(floorf(acc[j]) * INV_PV1F);
  }

  // ---- 2nd-smallest over f, update, clip (valid rows live in lanes 0-15) --
  if (lane < 16) {
    const float ur = (float)urate[0];
    const int gy = ty0 + py;
    const int gx = tx0 + px;
#pragma unroll
    for (int o = 0; o < 8; ++o) {
      float v0 = val[0][o], v1 = val[1][o], v2 = val[2][o], v3 = val[3][o];
      float lo1 = fminf(v0, v1), hi1 = fmaxf(v0, v1);
      float lo2 = fminf(v2, v3), hi2 = fmaxf(v2, v3);
      float z3 = fminf(fmaxf(lo1, lo2), fminf(hi1, hi2));  // sort[1] of 4
      size_t idx = ((size_t)(b * 8 + o) * 256 + gy) * 256 + gx;
      float xv = xin[idx];
      out[idx] = fminf(fmaxf(xv + z3 * ur, 0.0f), 1.0f);
    }
  }
}

extern "C" void kernel_launch(void* const* d_in, const int* in_sizes, int n_in,
                              void* d_out, int out_size, void* d_ws, size_t ws_size,
                              hipStream_t stream) {
  const float* x       = (const float*)d_in[0];
  const float* kernels = (const float*)d_in[1];
  const float* biases  = (const float*)d_in[2];
  const float* W1      = (const float*)d_in[3];
  const float* W2      = (const float*)d_in[4];
  const float* W3      = (const float*)d_in[5];
  const float* W4      = (const float*)d_in[6];
  const int*   urate   = (const int*)d_in[7];
  float* ws  = (float*)d_ws;
  float* outp = (float*)d_out;

  ca_prep<<<1, 256, 0, stream>>>(kernels, W1, W2, W3, W4, ws);

  dim3 grid(256 / 16, 256 / 8, 8);   // (x tiles, y tiles, batch)
  ca_main<<<grid, 256, 0, stream>>>(x, biases, ws, urate, outp);
}